// ResSS2D_89335319757371
// MI455X (gfx1250) — compile-verified
//
#include <hip/hip_runtime.h>
#include <math.h>

// ---- problem constants (from setup_inputs) ----
constexpr int B_  = 4;
constexpr int C_  = 96;     // dims
constexpr int H_  = 64;
constexpr int W_  = 64;
constexpr int L_  = 4096;   // H*W
constexpr int D_  = 192;    // Din
constexpr int R_  = 6;      // dt_rank
constexpr int K_  = 4;      // scan directions
constexpr int M_  = 384;    // mlp hidden
constexpr int P_  = B_ * L_; // 16384 pixels

typedef __attribute__((ext_vector_type(2))) float v2f;
typedef __attribute__((ext_vector_type(8))) float v8f;

// fp32 WMMA 16x16x4: D = A(16x4) * B(4x16) + C  -- keeps full fp32 precision
__device__ inline v8f wmma4(v2f a, v2f b, v8f c) {
  return __builtin_amdgcn_wmma_f32_16x16x4_f32(
      /*neg_a=*/false, a, /*neg_b=*/false, b,
      /*c_mod=*/(short)0, c, /*reuse_a=*/false, /*reuse_b=*/false);
}

__device__ inline float softplusf(float x) {
  return (x > 20.f) ? x : log1pf(expf(x));
}
__device__ inline float siluf(float x) {
  return x / (1.f + expf(-x));
}
__device__ inline float gelu_exactf(float x) {
  return 0.5f * x * (1.f + erff(x * 0.70710678118654752440f));
}

// direction mapping: xs[k][l] = uc[map(k,l)]  (cross-merge uses the same map)
__device__ inline int map_dir(int k, int l) {
  if (k == 0) return l;
  if (k == 1) return ((l & 63) << 6) | (l >> 6);       // transpose (h,w)
  if (k == 2) return (L_ - 1) - l;                     // reversed row-major
  int l2 = (L_ - 1) - l;                               // k == 3
  return ((l2 & 63) << 6) | (l2 >> 6);
}

// ---- CDNA5 async global->LDS staging (ASYNCcnt path) ----
__device__ inline unsigned lds_addr_of(const void* p) {
  return (unsigned)(unsigned long long)
      (__attribute__((address_space(3))) const void*)p;
}
__device__ inline void async_ld_b128(unsigned lds_byte_off, const float* g) {
  asm volatile("global_load_async_to_lds_b128 %0, %1, off"
               :: "v"(lds_byte_off), "v"(g) : "memory");
}
__device__ inline void wait_async0() {
  asm volatile("s_wait_asynccnt 0x0" ::: "memory");
}

// ============================================================
// K1: LayerNorm(norm1) + in_proj (96 -> 192) via WMMA fp32
//     128 threads = 4 waves; tile = 16 pixels; each wave does 3 N-tiles
//     x tile staged to LDS via global_load_async_to_lds_b128
// ============================================================
__global__ void k1_ln_inproj(const float* __restrict__ x,
                             const float* __restrict__ g1,
                             const float* __restrict__ b1,
                             const float* __restrict__ Wip,   // (192,96)
                             float* __restrict__ u_buf) {     // (P,192)
  __shared__ __align__(16) float raw[16 * 96];   // [c*16 + r]
  __shared__ float aln[16 * 96];                 // [r*96 + c], normalized
  __shared__ float mu[16], iv[16];
  const int tid = threadIdx.x;
  const int pg0 = blockIdx.x * 16;
  const int b = pg0 >> 12;
  const int pix0 = pg0 & 4095;

  if (tid == 0) __builtin_prefetch(Wip, 0, 1);

  // async stage: 96 channels x 4 chunks of 16B, contiguous in pixel r
  {
    const unsigned base = lds_addr_of(raw);
    for (int q = tid; q < 96 * 4; q += 128) {
      int c = q >> 2, r4 = (q & 3) << 2;
      async_ld_b128(base + (unsigned)((c * 16 + r4) * 4),
                    x + (size_t)(b * 96 + c) * 4096 + pix0 + r4);
    }
    wait_async0();
  }
  __syncthreads();

  if (tid < 16) {
    float s = 0.f;
    for (int c = 0; c < 96; ++c) s += raw[c * 16 + tid];
    float m = s * (1.f / 96.f);
    float v = 0.f;
    for (int c = 0; c < 96; ++c) { float d = raw[c * 16 + tid] - m; v += d * d; }
    mu[tid] = m;
    iv[tid] = rsqrtf(v * (1.f / 96.f) + 1e-5f);
  }
  __syncthreads();
  for (int idx = tid; idx < 16 * 96; idx += 128) {
    int c = idx >> 4, r = idx & 15;
    aln[r * 96 + c] = (raw[idx] - mu[r]) * iv[r] * g1[c] + b1[c];
  }
  __syncthreads();

  const int lane = tid & 31, wv = tid >> 5;
  const int mrow = lane & 15;
  const int kb = (lane >> 4) << 1;     // 0 or 2
  const int mb = (lane >> 4) << 3;     // D row offset 0 or 8

  #pragma unroll
  for (int t = 0; t < 3; ++t) {
    const int n0 = (wv * 3 + t) * 16;
    v8f acc = {};
    #pragma unroll
    for (int kk = 0; kk < 96; kk += 4) {
      v2f a, bf;
      a.x = aln[mrow * 96 + kk + kb];
      a.y = aln[mrow * 96 + kk + kb + 1];
      const float* wp = Wip + (n0 + mrow) * 96 + kk + kb;
      bf.x = wp[0];
      bf.y = wp[1];
      acc = wmma4(a, bf, acc);
    }
    #pragma unroll
    for (int r = 0; r < 8; ++r)
      u_buf[(pg0 + mb + r) * 192 + n0 + mrow] = acc[r];
  }
}

// ============================================================
// K2: depthwise 3x3 conv (SAME) + SiLU;  u_buf(P,192) -> uc(B,192,L)
//     one pixel per 192-thread block: all 9 taps read coalesced rows
// ============================================================
__global__ void k2_conv_silu(const float* __restrict__ u_buf,
                             const float* __restrict__ cw,   // (192,1,3,3)
                             float* __restrict__ uc) {
  const int d = threadIdx.x;
  const int pg = blockIdx.x;           // 0..P-1
  const int b = pg >> 12;
  const int l = pg & 4095;
  const int h = l >> 6, w = l & 63;
  float s = 0.f;
  #pragma unroll
  for (int dh = -1; dh <= 1; ++dh) {
    int hh = h + dh;
    if (hh < 0 || hh >= 64) continue;
    #pragma unroll
    for (int dw = -1; dw <= 1; ++dw) {
      int ww = w + dw;
      if (ww < 0 || ww >= 64) continue;
      s += u_buf[(b * 4096 + hh * 64 + ww) * 192 + d] *
           cw[d * 9 + (dh + 1) * 3 + (dw + 1)];
    }
  }
  uc[(b * 192 + d) * 4096 + l] = siluf(s);
}

// ============================================================
// K3: x_proj  ->  dbl(B,K,8,L);  xs never materialized
// grid (L/256, K, B)
// ============================================================
__global__ void k3_xproj(const float* __restrict__ uc,
                         const float* __restrict__ xw,   // (K,8,192)
                         float* __restrict__ dbl) {
  const int l = blockIdx.x * 256 + threadIdx.x;
  const int k = blockIdx.y;
  const int b = blockIdx.z;
  const int sl = map_dir(k, l);
  const float* wrow = xw + k * 8 * 192;
  float acc[8] = {0.f, 0.f, 0.f, 0.f, 0.f, 0.f, 0.f, 0.f};
  for (int d = 0; d < 192; ++d) {
    float v = uc[(b * 192 + d) * 4096 + sl];
    #pragma unroll
    for (int c = 0; c < 8; ++c) acc[c] += v * wrow[c * 192 + d];
  }
  #pragma unroll
  for (int c = 0; c < 8; ++c)
    dbl[(((b * 4 + k) * 8) + c) * 4096 + l] = acc[c];
}

// ============================================================
// K4: dt_proj + softplus + selective scan (N=1) + cross-merge
// grid (D, B); 256 threads; one block owns the whole (b,d) slice:
// all 4 directions accumulated into a 16KB LDS buffer (no global atomics)
// h_l = a_l * h_{l-1} + bb_l ;  y = h*C + D*x
// ============================================================
__global__ void k4_scan(const float* __restrict__ uc,
                        const float* __restrict__ dbl,
                        const float* __restrict__ dtw,   // (K,192,6)
                        const float* __restrict__ dtb,   // (K,192)
                        const float* __restrict__ alog,  // (768,1)
                        const float* __restrict__ Ds,    // (768,)
                        float* __restrict__ ym) {        // (B,L,192)
  __shared__ float acc[4096];
  __shared__ float sA[256], sB[256];
  const int tid = threadIdx.x;
  const int d = blockIdx.x, b = blockIdx.y;
  const float* ucd = uc + (b * 192 + d) * 4096;

  for (int i = tid; i < 4096; i += 256) acc[i] = 0.f;
  __syncthreads();

  const int base_l = tid * 16;
  for (int k = 0; k < 4; ++k) {
    const int kd = k * 192 + d;
    const float* wv = dtw + kd * 6;
    const float w0 = wv[0], w1 = wv[1], w2 = wv[2],
                w3 = wv[3], w4 = wv[4], w5 = wv[5];
    const float bd = dtb[kd];
    const float Aelem = -expf(alog[kd]);
    const float Dd = Ds[kd];
    const float* dblb = dbl + (b * 4 + k) * 8 * 4096;

    float a_arr[16], b_arr[16];
    float pa = 1.f, pb = 0.f;
    #pragma unroll
    for (int i = 0; i < 16; ++i) {
      int l = base_l + i;
      float s = dblb[0 * 4096 + l] * w0 + dblb[1 * 4096 + l] * w1 +
                dblb[2 * 4096 + l] * w2 + dblb[3 * 4096 + l] * w3 +
                dblb[4 * 4096 + l] * w4 + dblb[5 * 4096 + l] * w5 + bd;
      float dt = softplusf(s);
      float xv = ucd[map_dir(k, l)];
      float a = expf(dt * Aelem);
      float bb = dt * dblb[6 * 4096 + l] * xv;
      a_arr[i] = a; b_arr[i] = bb;
      pb = a * pb + bb;
      pa = a * pa;
    }
    sA[tid] = pa; sB[tid] = pb;
    __syncthreads();
    // inclusive Hillis-Steele scan of affine maps (a,b)
    for (int off = 1; off < 256; off <<= 1) {
      float ca = 1.f, cb = 0.f;
      if (tid >= off) { ca = sA[tid - off]; cb = sB[tid - off]; }
      float oa = sA[tid], ob = sB[tid];
      __syncthreads();
      sA[tid] = oa * ca;
      sB[tid] = oa * cb + ob;
      __syncthreads();
    }
    float h = (tid == 0) ? 0.f : sB[tid - 1];   // h before this chunk
    #pragma unroll
    for (int i = 0; i < 16; ++i) {
      int l = base_l + i;
      h = a_arr[i] * h + b_arr[i];
      float Cv = dblb[7 * 4096 + l];
      int sl = map_dir(k, l);
      float xv = ucd[sl];
      acc[sl] += h * Cv + Dd * xv;   // lp == sl; bijective per k -> race-free
    }
    __syncthreads();
  }

  for (int i = tid; i < 4096; i += 256)
    ym[(size_t)(b * 4096 + i) * 192 + d] = acc[i];
}

// ============================================================
// K5: LayerNorm(out_norm) + out_proj (192 -> 96, WMMA) + residual z0
// 192 threads = 6 waves; tile 16 pixels; one 16-wide N-tile per wave
// ym tile staged to LDS asynchronously (row-major, contiguous rows)
// ============================================================
__global__ void k5_outproj(const float* __restrict__ ym,
                           const float* __restrict__ og,
                           const float* __restrict__ ob,
                           const float* __restrict__ Wout,  // (96,192)
                           const float* __restrict__ x,
                           float* __restrict__ z_buf) {     // (P,96)
  __shared__ __align__(16) float raw[16 * 192];  // [r*192 + c]
  __shared__ float aln[16 * 192];                // [r*192 + c]
  __shared__ float mu[16], iv[16];
  const int tid = threadIdx.x;
  const int pg0 = blockIdx.x * 16;
  const int b = pg0 >> 12;
  const int pix0 = pg0 & 4095;

  {
    const unsigned base = lds_addr_of(raw);
    for (int q = tid; q < 16 * 48; q += 192) {
      int r = q / 48, c4 = (q % 48) << 2;
      async_ld_b128(base + (unsigned)((r * 192 + c4) * 4),
                    ym + (size_t)(pg0 + r) * 192 + c4);
    }
    wait_async0();
  }
  __syncthreads();

  if (tid < 16) {
    const float* row = raw + tid * 192;
    float s = 0.f;
    for (int c = 0; c < 192; ++c) s += row[c];
    float m = s * (1.f / 192.f);
    float v = 0.f;
    for (int c = 0; c < 192; ++c) { float d = row[c] - m; v += d * d; }
    mu[tid] = m;
    iv[tid] = rsqrtf(v * (1.f / 192.f) + 1e-5f);
  }
  __syncthreads();
  {
    float gv = og[tid], bv = ob[tid];
    #pragma unroll
    for (int r = 0; r < 16; ++r)
      aln[r * 192 + tid] = (raw[r * 192 + tid] - mu[r]) * iv[r] * gv + bv;
  }
  __syncthreads();

  const int lane = tid & 31, wv = tid >> 5;
  const int mrow = lane & 15;
  const int kb = (lane >> 4) << 1;
  const int mb = (lane >> 4) << 3;
  const int n0 = wv * 16;
  v8f acc = {};
  #pragma unroll
  for (int kk = 0; kk < 192; kk += 4) {
    v2f a, bf;
    a.x = aln[mrow * 192 + kk + kb];
    a.y = aln[mrow * 192 + kk + kb + 1];
    const float* wp = Wout + (n0 + mrow) * 192 + kk + kb;
    bf.x = wp[0];
    bf.y = wp[1];
    acc = wmma4(a, bf, acc);
  }
  #pragma unroll
  for (int r = 0; r < 8; ++r) {
    int row = mb + r, col = n0 + mrow;
    z_buf[(pg0 + row) * 96 + col] =
        x[(b * 96 + col) * 4096 + pix0 + row] + acc[r];
  }
}

// ============================================================
// K6: LayerNorm(norm2) + fc1(WMMA) + exact GELU + fc2(WMMA)
//     + MLP residual + outer residual -> out (B,C,H,W)
// 192 threads = 6 waves; tile 16 pixels; z tile staged asynchronously
// ============================================================
__global__ void k6_mlp(const float* __restrict__ z_buf,
                       const float* __restrict__ g2,
                       const float* __restrict__ b2,
                       const float* __restrict__ W1,  // (384,96)
                       const float* __restrict__ bb1, // (384,)
                       const float* __restrict__ W2,  // (96,384)
                       const float* __restrict__ bb2, // (96,)
                       const float* __restrict__ x,
                       float* __restrict__ out) {
  __shared__ __align__(16) float raw[16 * 96];   // [r*96 + c]
  __shared__ float Ms[16 * 96];                  // [r*96 + c] normalized
  __shared__ float Hs[16 * 384];                 // [r*384 + n] gelu(fc1)
  __shared__ float mu[16], iv[16];
  const int tid = threadIdx.x;
  const int pg0 = blockIdx.x * 16;
  const int b = pg0 >> 12;
  const int pix0 = pg0 & 4095;

  {
    const unsigned base = lds_addr_of(raw);
    for (int q = tid; q < 16 * 24; q += 192) {
      int r = q / 24, c4 = (q % 24) << 2;
      async_ld_b128(base + (unsigned)((r * 96 + c4) * 4),
                    z_buf + (size_t)(pg0 + r) * 96 + c4);
    }
    wait_async0();
  }
  __syncthreads();

  if (tid < 16) {
    const float* row = raw + tid * 96;
    float s = 0.f;
    for (int c = 0; c < 96; ++c) s += row[c];
    float m = s * (1.f / 96.f);
    float v = 0.f;
    for (int c = 0; c < 96; ++c) { float d = row[c] - m; v += d * d; }
    mu[tid] = m;
    iv[tid] = rsqrtf(v * (1.f / 96.f) + 1e-5f);
  }
  __syncthreads();
  for (int idx = tid; idx < 16 * 96; idx += 192) {
    int r = idx / 96, c = idx - r * 96;
    Ms[idx] = (raw[idx] - mu[r]) * iv[r] * g2[c] + b2[c];
  }
  __syncthreads();

  const int lane = tid & 31, wv = tid >> 5;
  const int mrow = lane & 15;
  const int kb = (lane >> 4) << 1;
  const int mb = (lane >> 4) << 3;

  // fc1: 384 outs = 24 tiles over 6 waves -> 4 tiles each
  #pragma unroll
  for (int t = 0; t < 4; ++t) {
    const int n0 = (wv * 4 + t) * 16;
    v8f acc = {};
    #pragma unroll
    for (int kk = 0; kk < 96; kk += 4) {
      v2f a, bf;
      a.x = Ms[mrow * 96 + kk + kb];
      a.y = Ms[mrow * 96 + kk + kb + 1];
      const float* wp = W1 + (n0 + mrow) * 96 + kk + kb;
      bf.x = wp[0];
      bf.y = wp[1];
      acc = wmma4(a, bf, acc);
    }
    #pragma unroll
    for (int r = 0; r < 8; ++r) {
      int row = mb + r, col = n0 + mrow;
      Hs[row * 384 + col] = gelu_exactf(acc[r] + bb1[col]);
    }
  }
  __syncthreads();

  // fc2: 96 outs = 6 tiles, one per wave; K = 384
  {
    const int n0 = wv * 16;
    v8f acc = {};
    #pragma unroll
    for (int kk = 0; kk < 384; kk += 4) {
      v2f a, bf;
      a.x = Hs[mrow * 384 + kk + kb];
      a.y = Hs[mrow * 384 + kk + kb + 1];
      const float* wp = W2 + (n0 + mrow) * 384 + kk + kb;
      bf.x = wp[0];
      bf.y = wp[1];
      acc = wmma4(a, bf, acc);
    }
    #pragma unroll
    for (int r = 0; r < 8; ++r) {
      int row = mb + r, col = n0 + mrow;
      float zres = z_buf[(pg0 + row) * 96 + col];
      float xv = x[(b * 96 + col) * 4096 + pix0 + row];
      out[(b * 96 + col) * 4096 + pix0 + row] = xv + zres + acc[r] + bb2[col];
    }
  }
}

// ============================================================
// host launcher
// ============================================================
extern "C" void kernel_launch(void* const* d_in, const int* in_sizes, int n_in,
                              void* d_out, int out_size, void* d_ws, size_t ws_size,
                              hipStream_t stream) {
  const float* x      = (const float*)d_in[0];
  const float* n1g    = (const float*)d_in[1];
  const float* n1b    = (const float*)d_in[2];
  const float* Wip    = (const float*)d_in[3];
  const float* convw  = (const float*)d_in[4];
  const float* xpw    = (const float*)d_in[5];
  const float* dtw    = (const float*)d_in[6];
  const float* dtb    = (const float*)d_in[7];
  const float* alog   = (const float*)d_in[8];
  const float* Ds     = (const float*)d_in[9];
  const float* ong    = (const float*)d_in[10];
  const float* onb    = (const float*)d_in[11];
  const float* Wout   = (const float*)d_in[12];
  const float* n2g    = (const float*)d_in[13];
  const float* n2b    = (const float*)d_in[14];
  const float* W1     = (const float*)d_in[15];
  const float* b1v    = (const float*)d_in[16];
  const float* W2     = (const float*)d_in[17];
  const float* b2v    = (const float*)d_in[18];
  float* out = (float*)d_out;

  float* ws = (float*)d_ws;
  float* u_buf = ws;                               // P*192
  float* uc    = u_buf + (size_t)P_ * 192;         // P*192 (channel-major)
  float* dbl   = uc + (size_t)P_ * 192;            // B*K*8*L
  float* ym    = dbl + (size_t)B_ * K_ * 8 * L_;   // P*192
  float* z_buf = ym + (size_t)P_ * 192;            // P*96

  k1_ln_inproj<<<P_ / 16, 128, 0, stream>>>(x, n1g, n1b, Wip, u_buf);
  k2_conv_silu<<<P_, 192, 0, stream>>>(u_buf, convw, uc);
  k3_xproj<<<dim3(L_ / 256, K_, B_), 256, 0, stream>>>(uc, xpw, dbl);
  k4_scan<<<dim3(D_, B_), 256, 0, stream>>>(uc, dbl, dtw, dtb, alog, Ds, ym);
  k5_outproj<<<P_ / 16, 192, 0, stream>>>(ym, ong, onb, Wout, x, z_buf);
  k6_mlp<<<P_ / 16, 192, 0, stream>>>(z_buf, n2g, n2b, W1, b1v, W2, b2v, x, out);
}